// MoEProjectionLayer_15659450761984
// MI455X (gfx1250) — compile-verified
//
#include <hip/hip_runtime.h>
#include <hip/hip_bf16.h>
#include <math.h>

#define B_ 8
#define K_ 4
#define D_ 384
#define L_ 4096
#define E_ 8
#define C_ 16   // DT_RANK

typedef __attribute__((ext_vector_type(2))) float v2f;
typedef __attribute__((ext_vector_type(8))) float v8f;

// ---------------- Kernel 1: mh[b,d] = mean over (k,l) of xs ----------------
__global__ void moe_mean_kernel(const float* __restrict__ xs, float* __restrict__ mh) {
    const int bd = blockIdx.x;            // b*D + d
    const int b  = bd / D_;
    const int d  = bd % D_;
    const int t  = threadIdx.x;
    float s = 0.f;
    for (int k = 0; k < K_; ++k) {
        const float4* p = (const float4*)(xs + ((size_t)(b * K_ + k) * D_ + d) * L_);
        for (int i = t; i < L_ / 4; i += 256) {
            float4 v = p[i];
            s += v.x + v.y + v.z + v.w;
        }
    }
    __shared__ float red[256];
    red[t] = s;
    __syncthreads();
    for (int off = 128; off > 0; off >>= 1) {
        if (t < off) red[t] += red[t + off];
        __syncthreads();
    }
    if (t == 0) mh[bd] = red[0] * (1.0f / (K_ * L_));
}

// ---------------- Kernel 2: noisy top-k gate + z_loss (tiny) ----------------
__global__ void moe_gate_kernel(const float* __restrict__ mh,
                                const int* __restrict__ mis_mask,
                                const float* __restrict__ route_w,
                                const float* __restrict__ route_b,
                                const float* __restrict__ noise_w,
                                const float* __restrict__ noise_b,
                                const float* __restrict__ noise,
                                float* __restrict__ gate,
                                float* __restrict__ zloss_out) {
    const int t = threadIdx.x;            // 64 threads: (b,e) pairs
    const int b = t >> 3, e = t & 7;
    __shared__ float s_log[64], s_nv[64], s_noisy[64], s_l2[8];
    float rl = route_b[e], nl = noise_b[e];
    const float* mhb = mh + b * D_;
    const float* rw  = route_w + e * D_;
    const float* nw  = noise_w + e * D_;
    for (int d = 0; d < D_; ++d) { float x = mhb[d]; rl += x * rw[d]; nl += x * nw[d]; }
    // softplus, numerically stable
    float sp = fmaxf(nl, 0.f) + log1pf(expf(-fabsf(nl)));
    s_log[t] = rl;
    s_nv[t]  = noise[b * E_ + e] * sp;
    __syncthreads();
    const int rb = b * 8;
    float m1 = -1e30f, m2 = -1e30f;
    for (int j = 0; j < 8; ++j) { m1 = fmaxf(m1, s_log[rb + j]); m2 = fmaxf(m2, s_nv[rb + j]); }
    float z1 = 0.f, z2 = 0.f;
    for (int j = 0; j < 8; ++j) { z1 += expf(s_log[rb + j] - m1); z2 += expf(s_nv[rb + j] - m2); }
    float noisy = expf(rl - m1) / z1 + expf(s_nv[t] - m2) / z2;
    s_noisy[t] = noisy;
    __syncthreads();
    const int kk = mis_mask[b];
    // sparse logits (rank < k keeps value, else 0), softmax over them
    float mx = -1e30f;
    float sv[8];
    for (int j = 0; j < 8; ++j) {
        float vj = s_noisy[rb + j];
        int rank = 0;
        for (int i2 = 0; i2 < 8; ++i2) {
            float vi = s_noisy[rb + i2];
            rank += (vi > vj) || (vi == vj && i2 < j);   // stable-descending rank
        }
        float s = (rank < kk) ? vj : 0.f;
        sv[j] = s;
        mx = fmaxf(mx, s);
    }
    float zs = 0.f;
    for (int j = 0; j < 8; ++j) zs += expf(sv[j] - mx);
    gate[t] = expf(sv[e] - mx) / zs;
    // z_loss = mean_b logsumexp(noisy_b)^2
    if (e == 0) {
        float m3 = -1e30f;
        for (int j = 0; j < 8; ++j) m3 = fmaxf(m3, s_noisy[rb + j]);
        float z3 = 0.f;
        for (int j = 0; j < 8; ++j) z3 += expf(s_noisy[rb + j] - m3);
        float lse = m3 + logf(z3);
        s_l2[b] = lse * lse;
    }
    __syncthreads();
    if (t == 0) {
        float acc = 0.f;
        for (int j = 0; j < 8; ++j) acc += s_l2[j];
        *zloss_out = acc * (1.0f / 8.0f);
    }
}

// ---------------- Kernel 3: Wc[b,k,c,d] = sum_e gate[b,e] * W[e,k,c,d] ----------------
__global__ void moe_wmix_kernel(const float* __restrict__ W, const float* __restrict__ gate,
                                float* __restrict__ Wc) {
    const int id = blockIdx.x;            // (b*K + k)*16 + c, 512 blocks
    const int b  = id / (K_ * C_);
    const int kc = id % (K_ * C_);
    const int d  = threadIdx.x;           // 384 threads
    float acc = 0.f;
    for (int e = 0; e < E_; ++e)
        acc += gate[b * E_ + e] * W[((size_t)e * K_ * C_ + kc) * D_ + d];
    Wc[(size_t)id * D_ + d] = acc;
}

// ---------------- Kernel 4: per-(b,k) GEMM out[16xL] = Wc[16xD] @ xs[DxL] via WMMA ----------------
// One wave computes a 16x64 output strip: 4 accumulator tiles sharing each A fragment.
// A (16x4 f32): lane m=lane%16 holds M=m; v0=K(0|2), v1=K(1|3) -> contiguous float2 from Wc row m.
// B (4x16 f32): v0 lanes0-15=K0 / lanes16-31=K2; v1=K1/K3; N=lane%16 -> coalesced xs rows.
// C/D (16x16):  VGPR r holds rows r (lanes 0-15) and r+8 (lanes 16-31).
__global__ void moe_gemm_wmma_kernel(const float* __restrict__ xs,
                                     const float* __restrict__ Wc,
                                     float* __restrict__ out) {
    const int lane  = threadIdx.x & 31;
    const int w     = blockIdx.x * (blockDim.x >> 5) + (threadIdx.x >> 5);  // global wave id
    const int bk    = w >> 6;            // 0..31 : (b,k) pair
    const int strip = w & 63;            // 0..63 : 64-column strip
    const int l0    = strip * 64;
    const int m  = lane & 15;
    const int kh = lane >> 4;            // 0: K half {0,1}, 1: K half {2,3}
    const float* xsp = xs + (size_t)bk * D_ * L_;
    const float* wp  = Wc + (size_t)bk * C_ * D_ + (size_t)m * D_;
    float* op = out + (size_t)bk * C_ * L_;

    v8f acc0 = {}, acc1 = {}, acc2 = {}, acc3 = {};
    for (int d0 = 0; d0 < D_; d0 += 4) {
        v2f a = *(const v2f*)(wp + d0 + 2 * kh);                       // A fragment (float2)
        const float* xr = xsp + (size_t)(d0 + 2 * kh) * L_ + l0 + m;
        if (d0 + 8 < D_) __builtin_prefetch(xr + 4 * (size_t)L_, 0, 3);
        v2f b0, b1, b2, b3;
        b0.x = xr[0];   b0.y = xr[L_];
        b1.x = xr[16];  b1.y = xr[L_ + 16];
        b2.x = xr[32];  b2.y = xr[L_ + 32];
        b3.x = xr[48];  b3.y = xr[L_ + 48];
        acc0 = __builtin_amdgcn_wmma_f32_16x16x4_f32(false, a, false, b0, (short)0, acc0, false, false);
        acc1 = __builtin_amdgcn_wmma_f32_16x16x4_f32(false, a, false, b1, (short)0, acc1, false, false);
        acc2 = __builtin_amdgcn_wmma_f32_16x16x4_f32(false, a, false, b2, (short)0, acc2, false, false);
        acc3 = __builtin_amdgcn_wmma_f32_16x16x4_f32(false, a, false, b3, (short)0, acc3, false, false);
    }
    const int col = l0 + m;
    for (int r = 0; r < 8; ++r) {
        float* orow = op + (size_t)(r + 8 * kh) * L_ + col;
        orow[0]  = acc0[r];
        orow[16] = acc1[r];
        orow[32] = acc2[r];
        orow[48] = acc3[r];
    }
}

extern "C" void kernel_launch(void* const* d_in, const int* in_sizes, int n_in,
                              void* d_out, int out_size, void* d_ws, size_t ws_size,
                              hipStream_t stream) {
    const float* xs       = (const float*)d_in[0];
    const int*   mis_mask = (const int*)d_in[1];
    const float* W        = (const float*)d_in[2];
    const float* route_w  = (const float*)d_in[3];
    const float* route_b  = (const float*)d_in[4];
    const float* noise_w  = (const float*)d_in[5];
    const float* noise_b  = (const float*)d_in[6];
    const float* noise    = (const float*)d_in[7];
    float* out = (float*)d_out;

    float* mh   = (float*)d_ws;            // B*D
    float* gate = mh + B_ * D_;            // B*E
    float* Wc   = gate + B_ * E_;          // B*K*16*D

    moe_mean_kernel<<<B_ * D_, 256, 0, stream>>>(xs, mh);
    moe_gate_kernel<<<1, 64, 0, stream>>>(mh, mis_mask, route_w, route_b,
                                          noise_w, noise_b, noise, gate,
                                          out + (size_t)B_ * K_ * C_ * L_);
    moe_wmix_kernel<<<B_ * K_ * C_, D_, 0, stream>>>(W, gate, Wc);
    // 32 (b,k) pairs * 64 strips = 2048 waves; 8 waves (256 threads) per block
    moe_gemm_wmma_kernel<<<(B_ * K_ * 64) / 8, 256, 0, stream>>>(xs, Wc, out);
}